// Model_87265145520641
// MI455X (gfx1250) — compile-verified
//
#include <hip/hip_runtime.h>
#include <hip/hip_bf16.h>

#define N_SAMPLES 131072
#define N_EVENTS  128
#define N_ATOMS   32
#define ATOM_SIZE 512
#define LATENT    16
#define TIME_DIM  17
#define LAYERS    7

typedef float v2f __attribute__((ext_vector_type(2)));
typedef float v8f __attribute__((ext_vector_type(8)));

// LDS layout in floats (total 71552 floats = 286208 bytes <= 320KB WGP LDS):
//  [0      .. 65536)  waveA (128 x 512)   -- tree ping-pong buffers aliased at its base
//      [0..2048) latA | [2048..4096) latB | [4096..8448) tmsA | [8448..12800) tmsB
//  [65536 .. 69632)  coef  (128 x 32)
//  [69632 .. 69760)  amp   (128)
//  [69760 .. 69888)  scale (128)
//  [69888 .. 70400)  win   (512)
//  [70400 .. 71424)  part  (1024)
//  [71424 .. 71552)  pos   (128, int)
#define SMEM_FLOATS 71552

extern "C" __global__ __launch_bounds__(1024, 1)
void synth_fused_kernel(const float* __restrict__ base_latent,
                        const float* __restrict__ to_time_W,
                        const float* __restrict__ split_W,
                        const float* __restrict__ split_b,
                        const float* __restrict__ atoms,
                        const float* __restrict__ to_atoms_W,
                        const float* __restrict__ to_atoms_b,
                        const float* __restrict__ to_amp_W,
                        const float* __restrict__ to_amp_b,
                        float* __restrict__ out)
{
    extern __shared__ float smem[];
    float* waveA  = smem;
    float* coef   = smem + 65536;
    float* ampS   = smem + 69632;
    float* scaleS = smem + 69760;
    float* winS   = smem + 69888;
    float* part   = smem + 70400;
    int*   posS   = (int*)(smem + 71424);

    const int tid = threadIdx.x;

    // ---- phase 0: zero output, build Hamming window, seed tree root ----
    for (int i = tid; i < N_SAMPLES; i += 1024) out[i] = 0.0f;
    for (int i = tid; i < ATOM_SIZE; i += 1024)
        winS[i] = 0.54f - 0.46f * cosf(6.28318530717958647692f * (float)i / (float)ATOM_SIZE);
    if (tid < LATENT)       smem[tid] = base_latent[tid];          // latA root
    if (tid < 2 * TIME_DIM) smem[4096 + tid] = 0.0f;               // tmsA root
    __syncthreads();

    // ---- phase 1: 7-layer binary latent tree + time-offset accumulation ----
    int cur = 0;
    for (int L = 0; L < LAYERS; ++L) {
        const int ne = 1 << L;
        const float* Wt = to_time_W + L * (2 * TIME_DIM * 2) * LATENT;  // (68,16)
        const float* Ws = split_W   + L * (2 * LATENT) * LATENT;        // (32,16)
        const float* Bs = split_b   + L * (2 * LATENT);
        const float* latC = smem + (cur ? 2048 : 0);
        float*       latN = smem + (cur ? 0 : 2048);
        const float* tmsC = smem + 4096 + (cur ? 4352 : 0);
        float*       tmsN = smem + 4096 + (cur ? 0 : 4352);
        const int total = ne * 100;   // per parent: 2 children x (16 latent + 34 time) outputs
        for (int w = tid; w < total; w += 1024) {
            const int e = w / 100;
            const int r = w - e * 100;
            const int c = r / 50;
            const int o = r - c * 50;
            const float* xv = latC + e * LATENT;
            if (o < LATENT) {
                const int row = c * LATENT + o;
                const float* wr = Ws + row * LATENT;
                float s = Bs[row];
                #pragma unroll
                for (int k = 0; k < LATENT; ++k) s += xv[k] * wr[k];
                latN[(e * 2 + c) * LATENT + o] = s;
            } else {
                const int d   = o - LATENT;                 // 0..33 = tdim*2 + comp
                const int row = c * (2 * TIME_DIM) + d;
                const float* wr = Wt + row * LATENT;
                float s = tmsC[e * (2 * TIME_DIM) + d];
                #pragma unroll
                for (int k = 0; k < LATENT; ++k) s += xv[k] * wr[k];
                tmsN[(e * 2 + c) * (2 * TIME_DIM) + d] = s;
            }
        }
        __syncthreads();
        cur ^= 1;
    }
    const float* latF = smem + (cur ? 2048 : 0);
    const float* tmsF = smem + 4096 + (cur ? 4352 : 0);

    // ---- phase 2: atom coefficients, amplitudes, Dirac positions ----
    for (int w = tid; w < N_EVENTS * N_ATOMS; w += 1024) {
        const int e = w >> 5, a = w & 31;
        const float* xv = latF + e * LATENT;
        const float* wr = to_atoms_W + a * LATENT;
        float s = to_atoms_b[a];
        #pragma unroll
        for (int k = 0; k < LATENT; ++k) s += xv[k] * wr[k];
        coef[e * N_ATOMS + a] = s;
    }
    if (tid < N_EVENTS) {
        const int e = tid;
        const float* xv = latF + e * LATENT;
        float s = to_amp_b[0];
        #pragma unroll
        for (int k = 0; k < LATENT; ++k) s += xv[k] * to_amp_W[k];
        ampS[e] = s;
        int p = 0;
        #pragma unroll
        for (int i = 0; i < TIME_DIM; ++i) {
            const float t0 = tmsF[e * (2 * TIME_DIM) + 2 * i + 0];
            const float t1 = tmsF[e * (2 * TIME_DIM) + 2 * i + 1];
            p = (p << 1) | ((t1 > t0) ? 1 : 0);   // argmax, first-index tie-break
        }
        posS[e] = p;
    }
    __syncthreads();

    // ---- phase 3: waveA[128x512] = coef[128x32] @ atoms[32x512], windowed ----
    // f32 WMMA 16x16x4; 8 M-tiles x 32 N-tiles over 32 waves, 8 K-steps each.
    {
        const int wv   = tid >> 5;
        const int lane = tid & 31;
        const int hh   = lane >> 4;   // half-wave selector
        const int lrow = lane & 15;
        for (int t = wv; t < 8 * 32; t += 32) {
            const int mt = t & 7;
            const int nt = t >> 3;
            v8f c = {0.f, 0.f, 0.f, 0.f, 0.f, 0.f, 0.f, 0.f};
            const float* arow = coef  + (mt * 16 + lrow) * N_ATOMS;
            const float* bcol = atoms + nt * 16 + lrow;
            #pragma unroll
            for (int kk = 0; kk < 8; ++kk) {
                const int k = kk * 4 + 2 * hh;
                v2f a, b;
                a.x = arow[k];
                a.y = arow[k + 1];
                b.x = bcol[(size_t)k * ATOM_SIZE];
                b.y = bcol[(size_t)(k + 1) * ATOM_SIZE];
                c = __builtin_amdgcn_wmma_f32_16x16x4_f32(
                        /*neg_a=*/false, a, /*neg_b=*/false, b,
                        /*c_mod=*/(short)0, c, /*reuse_a=*/false, /*reuse_b=*/false);
            }
            const float wvn = winS[nt * 16 + lrow];
            #pragma unroll
            for (int v = 0; v < 8; ++v) {
                waveA[(mt * 16 + v + 8 * hh) * ATOM_SIZE + nt * 16 + lrow] = c[v] * wvn;
            }
        }
    }
    __syncthreads();

    // ---- phase 4: per-event L2 norm -> scale = amp / (||a|| + 1e-8) ----
    {
        const int e   = tid >> 3;
        const int sub = tid & 7;
        float s = 0.f;
        const float* row = waveA + e * ATOM_SIZE + sub * 64;
        #pragma unroll
        for (int n = 0; n < 64; ++n) { const float v = row[n]; s += v * v; }
        part[tid] = s;
    }
    __syncthreads();
    if (tid < N_EVENTS) {
        float s = 0.f;
        #pragma unroll
        for (int i = 0; i < 8; ++i) s += part[tid * 8 + i];
        scaleS[tid] = ampS[tid] / (sqrtf(s) + 1e-8f);
    }
    __threadfence();          // zeroed out[] visible before atomics
    __syncthreads();

    // ---- phase 5: scatter-add each event's atom at its Dirac position ----
    for (int w = tid; w < N_EVENTS * ATOM_SIZE; w += 1024) {
        const int e = w >> 9;
        const int n = w & 511;
        const int tpos = posS[e] + n;
        if (tpos < N_SAMPLES)
            atomicAdd(&out[tpos], waveA[w] * scaleS[e]);
    }
}

extern "C" void kernel_launch(void* const* d_in, const int* in_sizes, int n_in,
                              void* d_out, int out_size, void* d_ws, size_t ws_size,
                              hipStream_t stream) {
    (void)in_sizes; (void)n_in; (void)out_size; (void)d_ws; (void)ws_size;
    synth_fused_kernel<<<dim3(1), dim3(1024), SMEM_FLOATS * sizeof(float), stream>>>(
        (const float*)d_in[0],  // base_latent (1,16)
        (const float*)d_in[1],  // to_time_W  (7,68,16)
        (const float*)d_in[2],  // split_W    (7,32,16)
        (const float*)d_in[3],  // split_b    (7,32)
        (const float*)d_in[4],  // atoms      (32,512)
        (const float*)d_in[5],  // to_atoms_W (32,16)
        (const float*)d_in[6],  // to_atoms_b (32,)
        (const float*)d_in[7],  // to_amp_W   (1,16)
        (const float*)d_in[8],  // to_amp_b   (1,)
        (float*)d_out);         // (1,1,131072) f32
}